// GCNLayer_55490977464423
// MI455X (gfx1250) — compile-verified
//
#include <hip/hip_runtime.h>
#include <math.h>

// GCN layer for MI455X (gfx1250, wave32).
// Pipeline: deg -> dis=rsqrt(deg) -> self-loop init of agg (in d_out) ->
//           atomic scatter of bidirectional messages -> FP32 WMMA GEMM
//           (out = relu(agg @ W^T + bias)), agg aliased with out (safe:
//           each row is read fully by its owning wave before being written).

typedef float v2f __attribute__((ext_vector_type(2)));
typedef float v8f __attribute__((ext_vector_type(8)));

#define D 128  // D_IN == D_OUT == 128

// ---------------- degree / norm ----------------

__global__ void k_init_deg(float* deg, int N) {
    int i = blockIdx.x * blockDim.x + threadIdx.x;
    if (i < N) deg[i] = 1.0f;  // self-loop contribution
}

__global__ void k_edge_deg(const long long* __restrict__ src,
                           const long long* __restrict__ dst,
                           float* deg, int E) {
    int e = blockIdx.x * blockDim.x + threadIdx.x;
    if (e < E) {
        // bidirectional: original edge hits dst, reversed edge hits src
        atomicAdd(&deg[(int)dst[e]], 1.0f);
        atomicAdd(&deg[(int)src[e]], 1.0f);
    }
}

__global__ void k_dis(const float* __restrict__ deg, float* dis, int N) {
    int i = blockIdx.x * blockDim.x + threadIdx.x;
    if (i < N) dis[i] = rsqrtf(fmaxf(deg[i], 1.0f));  // deg >= 1 always
}

// agg[i] = x[i] * dis[i]^2   (the self-loop message), one wave per node
__global__ void k_selfloop(const float* __restrict__ x,
                           const float* __restrict__ dis,
                           float* agg, int N) {
    int t = blockIdx.x * blockDim.x + threadIdx.x;
    int node = t >> 5;
    int lane = t & 31;  // lane owns 4 consecutive floats: 32*4 = 128
    if (node < N) {
        float ds = dis[node];
        float c = ds * ds;
        float4 v = ((const float4*)(x + (size_t)node * D))[lane];
        float4 r;
        r.x = v.x * c; r.y = v.y * c; r.z = v.z * c; r.w = v.w * c;
        ((float4*)(agg + (size_t)node * D))[lane] = r;
    }
}

// One wave per undirected edge; handles BOTH directions so the edge indices,
// dis values and coefficient are loaded once. 8 float atomics per lane.
// x/agg are L2-resident (51 MB each vs 192 MB L2), so atomics settle in L2.
__global__ void k_scatter(const long long* __restrict__ src,
                          const long long* __restrict__ dst,
                          const float* __restrict__ x,
                          const float* __restrict__ dis,
                          float* agg, int E) {
    int wid  = (blockIdx.x * blockDim.x + threadIdx.x) >> 5;
    int lane = threadIdx.x & 31;
    if (wid >= E) return;
    int s = (int)src[wid];
    int t = (int)dst[wid];
    float c = dis[s] * dis[t];
    float4 xs = ((const float4*)(x + (size_t)s * D))[lane];
    float4 xt = ((const float4*)(x + (size_t)t * D))[lane];
    float* at = agg + (size_t)t * D + lane * 4;
    float* as = agg + (size_t)s * D + lane * 4;
    atomicAdd(at + 0, xs.x * c);
    atomicAdd(at + 1, xs.y * c);
    atomicAdd(at + 2, xs.z * c);
    atomicAdd(at + 3, xs.w * c);
    atomicAdd(as + 0, xt.x * c);
    atomicAdd(as + 1, xt.y * c);
    atomicAdd(as + 2, xt.z * c);
    atomicAdd(as + 3, xt.w * c);
}

// ---------------- FP32 WMMA GEMM: out = relu(agg @ W^T + bias) ------------
// 256 threads = 8 wave32 per block; wave w owns rows [m0, m0+16).
//
// W is staged in 64 KB LDS in a PAIR-INTERLEAVED transposed layout so each
// lane's whole B fragment is one contiguous, aligned 8-byte LDS load:
//   for K-quad kq (k = 4*kq) and half h in {0,1}:
//     ldsB[ ((kq*2 + h)*128 + o)*2 + {0,1} ] = { W[o][4kq+h], W[o][4kq+h+2] }
// B frag (4x16 f32): lane l (h = l>>4, n = l&15) needs exactly
//   V0 = WT[k+h][o+n], V1 = WT[k+h+2][o+n]  ->  that float2.
// Consecutive lanes read consecutive float2s -> bank-conflict-free, and no
// VGPR repacking movs are needed before v_wmma.
//
// Per wave: 32 K-steps x 8 N-tiles = 256 v_wmma_f32_16x16x4_f32.
//
// Ragged last tile: A-row m only influences D-row m, and D rows >= N are
// guarded at the store, so out-of-range A rows are clamped in-bounds and
// loaded unconditionally (single straight global_load_b64 per K-step).

__global__ __launch_bounds__(256) void k_gemm(const float* aggIn,
                                              const float* __restrict__ W,
                                              const float* __restrict__ bias,
                                              float* outp, int N) {
    __shared__ float ldsB[D * D];  // 65536 bytes, pair-interleaved layout

    // cooperative stage: coalesced float4 global reads of W[o][4kq..4kq+3]
    const float4* W4 = (const float4*)W;
    for (int idx = threadIdx.x; idx < D * (D / 4); idx += blockDim.x) {
        int o  = idx >> 5;   // W row (output channel)
        int kq = idx & 31;   // K-quad
        float4 v = W4[idx];  // { K=4kq, 4kq+1, 4kq+2, 4kq+3 }
        v2f p0; p0.x = v.x; p0.y = v.z;  // half 0 pair {k, k+2}
        v2f p1; p1.x = v.y; p1.y = v.w;  // half 1 pair {k+1, k+3}
        *(v2f*)(ldsB + ((kq * 2 + 0) * D + o) * 2) = p0;
        *(v2f*)(ldsB + ((kq * 2 + 1) * D + o) * 2) = p1;
    }
    __syncthreads();

    int wave = threadIdx.x >> 5;
    int lane = threadIdx.x & 31;
    int half = lane >> 4;   // 0: lanes 0-15, 1: lanes 16-31
    int l16  = lane & 15;
    int m0 = blockIdx.x * 128 + wave * 16;
    if (m0 >= N) return;    // no further barriers after this point

    // clamp A row into bounds (garbage rows are never stored)
    int arow = m0 + l16;
    if (arow > N - 1) arow = N - 1;
    const float* Arow = aggIn + (size_t)arow * D + 2 * half;

    // per-lane base into the pair-interleaved B: (half, l16) fixed
    const float* Bbase = ldsB + (half * D + l16) * 2;

    v8f acc[8] = {};  // 8 N-tiles x 8 VGPRs f32 accumulators

    for (int kq = 0; kq < D / 4; ++kq) {
        // A frag 16x4 f32 (ISA layout): lanes 0-15 -> K={k,k+1},
        // lanes 16-31 -> K={k+2,k+3}; row = m0 + (lane&15).
        v2f a = *(const v2f*)(Arow + kq * 4);
        const float* Bk = Bbase + kq * (2 * D * 2);  // advance K-quad
#pragma unroll
        for (int t = 0; t < 8; ++t) {
            // one aligned 8B LDS load = the whole B fragment for tile t
            v2f b = *(const v2f*)(Bk + t * 32);
            acc[t] = __builtin_amdgcn_wmma_f32_16x16x4_f32(
                false, a, false, b, (short)0, acc[t], false, false);
        }
    }

    // epilogue: D layout -> VGPR r holds M = r + 8*half, N = lane&15
#pragma unroll
    for (int t = 0; t < 8; ++t) {
        int o = t * 16 + l16;
        float bv = bias[o];
#pragma unroll
        for (int r = 0; r < 8; ++r) {
            int row = m0 + r + 8 * half;
            if (row < N) {
                float v = acc[t][r] + bv;
                outp[(size_t)row * D + o] = v > 0.0f ? v : 0.0f;
            }
        }
    }
}

// ---------------- host side ----------------

extern "C" void kernel_launch(void* const* d_in, const int* in_sizes, int n_in,
                              void* d_out, int out_size, void* d_ws, size_t ws_size,
                              hipStream_t stream) {
    const float*     x    = (const float*)d_in[0];
    const long long* ei   = (const long long*)d_in[1];  // int64, shape (2, E)
    const float*     W    = (const float*)d_in[2];
    const float*     bias = (const float*)d_in[3];

    int N = in_sizes[0] / D;
    int E = in_sizes[1] / 2;
    const long long* src = ei;       // row 0
    const long long* dst = ei + E;   // row 1

    float* out = (float*)d_out;      // doubles as agg buffer
    float* deg = (float*)d_ws;       // N floats
    float* dis = deg + N;            // N floats  (0.8 MB total in ws)

    k_init_deg<<<(N + 255) / 256, 256, 0, stream>>>(deg, N);
    k_edge_deg<<<(E + 255) / 256, 256, 0, stream>>>(src, dst, deg, E);
    k_dis<<<(N + 255) / 256, 256, 0, stream>>>(deg, dis, N);

    // one wave per node: N*32 threads
    long long selfThreads = (long long)N * 32;
    k_selfloop<<<(int)((selfThreads + 255) / 256), 256, 0, stream>>>(x, dis, out, N);

    // one wave per undirected edge: E*32 threads
    long long scatThreads = (long long)E * 32;
    k_scatter<<<(int)((scatThreads + 255) / 256), 256, 0, stream>>>(src, dst, x, dis, out, E);

    // WMMA GEMM + bias + relu (agg aliased with out)
    k_gemm<<<(N + 127) / 128, 256, 0, stream>>>(out, W, bias, out, N);
}